// DHGLayer_47090021433857
// MI455X (gfx1250) — compile-verified
//
#include <hip/hip_runtime.h>
#include <hip/hip_bf16.h>

#define DD    128     // feature dim
#define SS    8       // hyperedges per vertex
#define KK    16      // members per hyperedge
#define ROWS  128     // SS*KK
#define HVC   9       // vertex-conv hidden
#define HEC   32      // edge-conv hidden
#define GSTR  132     // padded LDS row stride (floats): 132 mod 64 = 4 -> no bank conflicts

typedef float v2f __attribute__((ext_vector_type(2)));
typedef float v8f __attribute__((ext_vector_type(8)));

// ---------------------------------------------------------------------------
// Kernel A: per-vertex gather + vertex-conv attention pool (WMMA f32) +
// edge-conv attention pool. One block (128 threads, 4 wave32) per vertex.
//
// Vertex conv: hyperedge s is H_s(16x16) = G_s(16x128) @ W1pad(128x16),
// done as 32 chained V_WMMA_F32_16X16X4_F32 per hyperedge; wave w owns
// hyperedges 2w and 2w+1. Hidden dim padded 9->16 with zero columns, which
// self-neutralize through relu()*w2pad(=0).
// ---------------------------------------------------------------------------
__global__ __launch_bounds__(128) void dhg_pool_kernel(
    const int*   __restrict__ ids,
    const float* __restrict__ feats,
    const int*   __restrict__ adj,
    const float* __restrict__ vc_w1, const float* __restrict__ vc_b1,
    const float* __restrict__ vc_w2, const float* __restrict__ vc_b2,
    const float* __restrict__ ec_w1, const float* __restrict__ ec_b1,
    const float* __restrict__ ec_w2, const float* __restrict__ ec_b2,
    float*       __restrict__ xout)
{
    extern __shared__ float smem[];
    float* G   = smem;                 // ROWS*GSTR  gathered features
    float* HE  = G   + ROWS * GSTR;    // SS*GSTR    pooled hyperedge features
    float* w1t = HE  + SS * GSTR;      // 16*GSTR    vc_w1 transposed + padded
    float* b1s = w1t + 16 * GSTR;      // 16
    float* w2s = b1s + 16;             // 16
    float* eb1 = w2s + 16;             // 32
    float* ew2 = eb1 + 32;             // 32
    float* sc  = ew2 + 32;             // ROWS  raw scores -> softmax weights
    float* sc2 = sc  + ROWS;           // 8     hyperedge scores
    int*   nb  = (int*)(sc2 + 8);      // ROWS  neighbor indices

    const int tid  = threadIdx.x;
    const int bid  = blockIdx.x;
    const int lane = tid & 31;
    const int wv   = tid >> 5;         // wave 0..3
    const int mrow = lane & 15;        // fragment row (A) / col (B,C)
    const int hi   = lane >> 4;        // K-half select (ISA A/B frag layout)

    // stage vc_w1 transposed + zero-padded to 16 hidden cols: w1t[j][d]
    for (int i = tid; i < 16 * DD; i += 128) {
        const int j = i >> 7;          // hidden index 0..15
        const int d = i & (DD - 1);
        w1t[j * GSTR + d] = (j < HVC) ? vc_w1[d * HVC + j] : 0.0f;
    }
    if (tid < 16) {
        b1s[tid] = (tid < HVC) ? vc_b1[tid] : 0.0f;
        w2s[tid] = (tid < HVC) ? vc_w2[tid] : 0.0f;
    }
    if (tid < 32) { eb1[tid] = ec_b1[tid]; ew2[tid] = ec_w2[tid]; }

    const int vid = ids[bid];
    nb[tid] = adj[(size_t)vid * ROWS + tid];
    __syncthreads();

    // ---- coalesced gather: 128 rows x 128 floats into LDS ----
    #pragma unroll 4
    for (int r = 0; r < ROWS; ++r) {
        if (r + 4 < ROWS)
            __builtin_prefetch(&feats[(size_t)nb[r + 4] * DD + tid], 0, 0);
        G[r * GSTR + tid] = feats[(size_t)nb[r] * DD + tid];
    }
    __syncthreads();

    // ---- vertex conv via WMMA: H_s = G_s @ W1pad, then per-row scores ----
    const float b2v = vc_b2[0];
    #pragma unroll
    for (int p = 0; p < 2; ++p) {
        const int s = 2 * wv + p;
        v8f hc = {};
        #pragma unroll
        for (int kb = 0; kb < 32; ++kb) {
            const int k0 = kb * 4 + 2 * hi;
            // A: lane holds G[s*16+mrow][k0 .. k0+1]
            v2f a = *(const v2f*)&G[(s * KK + mrow) * GSTR + k0];
            // B: lane holds W1pad[k0][mrow], W1pad[k0+1][mrow] (contiguous in w1t)
            v2f b = *(const v2f*)&w1t[mrow * GSTR + k0];
            hc = __builtin_amdgcn_wmma_f32_16x16x4_f32(
                    false, a, false, b, (short)0, hc, false, false);
        }
        // score_m = b2 + sum_j relu(H[m][j]+b1[j])*w2[j]; lane col j = mrow
        const float b1v = b1s[mrow];
        const float w2v = w2s[mrow];
        #pragma unroll
        for (int r = 0; r < 8; ++r) {       // C layout: VGPR r -> row r (+8 if hi)
            float t = fmaxf(hc[r] + b1v, 0.0f) * w2v;
            #pragma unroll
            for (int mm = 8; mm >= 1; mm >>= 1) t += __shfl_xor(t, mm, 16);
            if (mrow == 0) sc[s * KK + r + 8 * hi] = t + b2v;
        }
    }
    __syncthreads();

    // ---- softmax over the 16 members of each hyperedge (sub-wave32) ----
    {
        float sco = sc[tid];
        float mx = sco;
        #pragma unroll
        for (int m = 8; m >= 1; m >>= 1) mx = fmaxf(mx, __shfl_xor(mx, m, 16));
        float e = __expf(sco - mx);
        float den = e;
        #pragma unroll
        for (int m = 8; m >= 1; m >>= 1) den += __shfl_xor(den, m, 16);
        sc[tid] = e / den;                  // own slot read->write, no hazard
    }
    __syncthreads();

    // ---- weighted member sum: thread owns column d = tid ----
    #pragma unroll
    for (int s = 0; s < SS; ++s) {
        float acc = 0.0f;
        #pragma unroll
        for (int k = 0; k < KK; ++k) {
            const int r = s * KK + k;
            acc = fmaf(sc[r], G[r * GSTR + tid], acc);
        }
        HE[s * GSTR + tid] = acc;
    }
    __syncthreads();

    // ---- edge conv: 256 (s,j) dot-products -> 2 per thread ----
    {
        const int j  = tid & 31;
        const int s1 = tid >> 5;        // 0..3
        const int s2 = s1 + 4;          // 4..7
        float a1 = eb1[j], a2 = eb1[j];
        for (int d = 0; d < DD; ++d) {
            const float w = ec_w1[d * HEC + j];   // coalesced, L2-hot
            a1 = fmaf(HE[s1 * GSTR + d], w, a1);
            a2 = fmaf(HE[s2 * GSTR + d], w, a2);
        }
        float t1 = fmaxf(a1, 0.0f) * ew2[j];
        float t2 = fmaxf(a2, 0.0f) * ew2[j];
        #pragma unroll
        for (int m = 16; m >= 1; m >>= 1) {
            t1 += __shfl_xor(t1, m, 32);
            t2 += __shfl_xor(t2, m, 32);
        }
        if ((tid & 31) == 0) {
            sc2[s1] = t1 + ec_b2[0];
            sc2[s2] = t2 + ec_b2[0];
        }
    }
    __syncthreads();

    // ---- softmax over 8 hyperedges (redundant per thread, cheap) ----
    float m8 = sc2[0];
    #pragma unroll
    for (int s = 1; s < SS; ++s) m8 = fmaxf(m8, sc2[s]);
    float es[SS], den2 = 0.0f;
    #pragma unroll
    for (int s = 0; s < SS; ++s) { es[s] = __expf(sc2[s] - m8); den2 += es[s]; }
    const float inv = 1.0f / den2;
    float x = 0.0f;
    #pragma unroll
    for (int s = 0; s < SS; ++s)
        x = fmaf(es[s] * inv, HE[s * GSTR + tid], x);

    xout[(size_t)bid * DD + tid] = x;
}

// ---------------------------------------------------------------------------
// Kernel B: Y = relu(X @ W + b) with V_WMMA_F32_16X16X4_F32 (exact f32 path).
// Block = 256 threads = 8 waves; block covers 16 rows x 128 cols; each wave
// owns one 16x16 tile (its column tile), K swept in 32 steps of 4.
// A tile is staged once in LDS and shared by all 8 waves.
// ---------------------------------------------------------------------------
__global__ __launch_bounds__(256) void dhg_fc_wmma_kernel(
    const float* __restrict__ X, const float* __restrict__ W,
    const float* __restrict__ bias, float* __restrict__ Y, int n)
{
    __shared__ float Xs[16 * GSTR];
    const int tid     = threadIdx.x;
    const int lane    = tid & 31;
    const int wv      = tid >> 5;          // wave -> column tile 0..7
    const int rowbase = blockIdx.x * 16;

    // stage 16 x 128 tile of X (clamp tail rows; stores stay in-bounds)
    for (int i = tid; i < 16 * (DD / 4); i += 256) {
        const int r  = i >> 5;             // DD/4 == 32
        const int c4 = i & 31;
        int row = rowbase + r; if (row >= n) row = n - 1;
        const float4 v = ((const float4*)(X + (size_t)row * DD))[c4];
        *((float4*)&Xs[r * GSTR + c4 * 4]) = v;
    }
    __syncthreads();

    const int m    = lane & 15;            // A row / B-C col within tile
    const int hi   = lane >> 4;            // K-half select (ISA A-frag layout)
    const int col0 = wv * 16;

    v8f c = {};
    #pragma unroll
    for (int kb = 0; kb < 32; ++kb) {
        const int k0 = kb * 4 + 2 * hi;
        // A: lane holds X[m][k0], X[m][k0+1]  (8B aligned in LDS)
        v2f a = *(const v2f*)&Xs[m * GSTR + k0];
        // B: lane holds W[k0][col0+m], W[k0+1][col0+m]
        v2f b;
        b.x = W[(size_t)k0 * DD + col0 + m];
        b.y = W[(size_t)(k0 + 1) * DD + col0 + m];
        c = __builtin_amdgcn_wmma_f32_16x16x4_f32(
                false, a, false, b, (short)0, c, false, false);
    }

    const float bv = bias[col0 + m];
    #pragma unroll
    for (int r = 0; r < 8; ++r) {          // C/D layout: VGPR r -> M=r (+8 if hi)
        const int row = rowbase + r + 8 * hi;
        if (row < n)
            Y[(size_t)row * DD + col0 + m] = fmaxf(c[r] + bv, 0.0f);
    }
}

// ---------------------------------------------------------------------------
extern "C" void kernel_launch(void* const* d_in, const int* in_sizes, int n_in,
                              void* d_out, int out_size, void* d_ws, size_t ws_size,
                              hipStream_t stream) {
    const int*   ids   = (const int*)  d_in[0];
    const float* feats = (const float*)d_in[1];
    const int*   adj   = (const int*)  d_in[2];
    const float* vc_w1 = (const float*)d_in[3];
    const float* vc_b1 = (const float*)d_in[4];
    const float* vc_w2 = (const float*)d_in[5];
    const float* vc_b2 = (const float*)d_in[6];
    const float* ec_w1 = (const float*)d_in[7];
    const float* ec_b1 = (const float*)d_in[8];
    const float* ec_w2 = (const float*)d_in[9];
    const float* ec_b2 = (const float*)d_in[10];
    const float* fc_w  = (const float*)d_in[11];
    const float* fc_b  = (const float*)d_in[12];

    const int n = in_sizes[0];             // N_IDS = 20000
    float* xws = (float*)d_ws;             // (n, 128) pooled features

    const size_t smemA =
        (size_t)(ROWS * GSTR + SS * GSTR + 16 * GSTR + 16 + 16 + 32 + 32 +
                 ROWS + 8) * sizeof(float) + ROWS * sizeof(int);

    dhg_pool_kernel<<<n, 128, smemA, stream>>>(
        ids, feats, adj, vc_w1, vc_b1, vc_w2, vc_b2,
        ec_w1, ec_b1, ec_w2, ec_b2, xws);

    const int nblk = (n + 15) / 16;
    dhg_fc_wmma_kernel<<<nblk, 256, 0, stream>>>(
        xws, fc_w, fc_b, (float*)d_out, n);
}